// SplineCNN_46059229283035
// MI455X (gfx1250) — compile-verified
//
#include <hip/hip_runtime.h>
#include <hip/hip_bf16.h>

typedef __attribute__((ext_vector_type(2))) float v2f;
typedef __attribute__((ext_vector_type(8))) float v8f;

#define CCH 64   // channels
#define NK  25   // 5x5 spline kernels

// ---------------------------------------------------------------------------
// XW GEMM via V_WMMA_F32_16X16X4_F32 with LDS staging.
// One block (128 thr = 4 wave32) owns a 16-node tile:
//   - stages A (16x64 f32, +2 pad floats/row so A-fragment reads hit distinct
//     banks: bank = (66*row + k) % 64) ONCE,
//   - loops k = 0..numK-1, staging each 64x64 W[k] (16 KB) with coalesced
//     float4 loads, then each wave computes one 16-col tile with 16 chained
//     fp32 WMMAs fed from LDS.
// Layouts per CDNA5 ISA 7.12.2 (A 16x4 / B 4x16 / C 16x16 striping).
// ---------------------------------------------------------------------------
__global__ __launch_bounds__(128) void xw_wmma_kernel(
    const float* __restrict__ X, const float* __restrict__ Wb, int numK,
    float* __restrict__ out, long outStride)
{
    __shared__ float As[16][CCH + 2];
    __shared__ float Ws[CCH][CCH];

    const int tid   = threadIdx.x;
    const int wave  = tid >> 5;
    const int lane  = tid & 31;
    const int half  = lane >> 4;      // selects K pair {0,1} vs {2,3}
    const int l16   = lane & 15;
    const int node0 = blockIdx.x * 16;
    const int col0  = wave * 16;

    // Stage A: 16 rows x 64 floats, coalesced 128-bit loads.
    {
        const float4* Xv = (const float4*)(X + (size_t)node0 * CCH);
#pragma unroll
        for (int i = tid; i < 16 * CCH / 4; i += 128) {
            const float4 v = Xv[i];
            const int row = i >> 4;          // 16 float4 per row
            const int c4  = (i & 15) * 4;
            As[row][c4 + 0] = v.x; As[row][c4 + 1] = v.y;
            As[row][c4 + 2] = v.z; As[row][c4 + 3] = v.w;
        }
    }

    for (int k = 0; k < numK; ++k) {
        const float4* Wv = (const float4*)(Wb + (size_t)k * CCH * CCH);
        __syncthreads();                      // Ws readers of iter k-1 done
#pragma unroll
        for (int i = tid; i < CCH * CCH / 4; i += 128) {
            const float4 v = Wv[i];
            const int row = i >> 4;
            const int c4  = (i & 15) * 4;
            Ws[row][c4 + 0] = v.x; Ws[row][c4 + 1] = v.y;
            Ws[row][c4 + 2] = v.z; Ws[row][c4 + 3] = v.w;
        }
        __syncthreads();                      // Ws (and first-iter As) visible

        v8f acc = {};
#pragma unroll
        for (int kb = 0; kb < CCH; kb += 4) {
            const int ka = kb + half * 2;
            v2f a = { As[l16][ka], As[l16][ka + 1] };
            v2f b = { Ws[ka][col0 + l16], Ws[ka + 1][col0 + l16] };
            acc = __builtin_amdgcn_wmma_f32_16x16x4_f32(
                false, a, false, b, (short)0, acc, false, false);
        }

        float* O = out + (size_t)k * CCH;
#pragma unroll
        for (int r = 0; r < 8; ++r) {
            const int m = half * 8 + r;
            O[(size_t)(node0 + m) * outStride + col0 + l16] = acc[r];
        }
    }
}

// ---------------------------------------------------------------------------
// Final linear: out = [x | x1 | x2] @ FW(192x64) + FB. K=192 WMMA loop with
// the A-source switched per 64-wide k block (concat-free). Single small
// launch; global loads hit L2-resident FW.
// ---------------------------------------------------------------------------
__global__ __launch_bounds__(128) void final_wmma_kernel(
    const float* __restrict__ x0, const float* __restrict__ x1,
    const float* __restrict__ x2, const float* __restrict__ FW,
    const float* __restrict__ FB, float* __restrict__ out)
{
    const int wave  = threadIdx.x >> 5;
    const int lane  = threadIdx.x & 31;
    const int half  = lane >> 4;
    const int l16   = lane & 15;
    const int node0 = blockIdx.x * 16;
    const int col0  = wave * 16;

    v8f acc = {};
#pragma unroll
    for (int kb = 0; kb < 3 * CCH; kb += 4) {
        const int ka = kb + half * 2;                   // even; stays in-block
        const float* S = (ka < CCH) ? x0 : ((ka < 2 * CCH) ? x1 : x2);
        const int kl = ka & (CCH - 1);
        v2f a = { S[(size_t)(node0 + l16) * CCH + kl],
                  S[(size_t)(node0 + l16) * CCH + kl + 1] };
        v2f b = { FW[(size_t)ka       * CCH + col0 + l16],
                  FW[(size_t)(ka + 1) * CCH + col0 + l16] };
        acc = __builtin_amdgcn_wmma_f32_16x16x4_f32(
            false, a, false, b, (short)0, acc, false, false);
    }
    const float bias = FB[col0 + l16];
#pragma unroll
    for (int r = 0; r < 8; ++r) {
        const int m = half * 8 + r;
        out[(size_t)(node0 + m) * CCH + col0 + l16] = acc[r] + bias;
    }
}

// ---------------------------------------------------------------------------
// Segment-max scratch init: sentinel 0u (strictly below every encoded float).
// ---------------------------------------------------------------------------
__global__ void init_agg_kernel(unsigned* __restrict__ agg, int n)
{
    int i = blockIdx.x * blockDim.x + threadIdx.x;
    if (i < n) agg[i] = 0u;
}

// Monotonic order-preserving float->uint encoding for atomic max.
__device__ __forceinline__ unsigned enc_f32(float f)
{
    unsigned u = __float_as_uint(f);
    return (u & 0x80000000u) ? ~u : (u | 0x80000000u);
}

// ---------------------------------------------------------------------------
// Edge stage: per (edge, channel). Degree-1 open B-spline basis (K=5, dim=2),
// gather 4 rows of XW[src], blend, atomic-max (u32, no-return) into AGG[dst].
// Deterministic: max is order-independent. This is the HBM-bound stage
// (~1 GB/layer of gather + atomic traffic -> ~45 us/layer at 23.3 TB/s).
// ---------------------------------------------------------------------------
__global__ __launch_bounds__(256) void edge_kernel(
    const float* __restrict__ XW, const int* __restrict__ ei,
    const float* __restrict__ attr, unsigned* __restrict__ agg, int E)
{
    const int e = blockIdx.x * 4 + (threadIdx.x >> 6);
    const int c = threadIdx.x & (CCH - 1);
    if (e >= E) return;

    const int src = ei[e];
    const int dst = ei[E + e];
    const float p0 = attr[2 * e];
    const float p1 = attr[2 * e + 1];

    const float v0 = p0 * 4.0f, v1 = p1 * 4.0f;
    const float l0 = floorf(v0), l1 = floorf(v1);
    const float f0 = v0 - l0,    f1 = v1 - l1;
    const int   i0 = (int)l0,    i1 = (int)l1;

    const float* base = XW + (size_t)src * NK * CCH + c;
    float msg = 0.0f;
#pragma unroll
    for (int s1 = 0; s1 < 2; ++s1) {
#pragma unroll
        for (int s2 = 0; s2 < 2; ++s2) {
            const float bs = (s1 ? f0 : (1.0f - f0)) * (s2 ? f1 : (1.0f - f1));
            int k0 = i0 + s1; k0 = k0 < 0 ? 0 : (k0 > 4 ? 4 : k0);
            int k1 = i1 + s2; k1 = k1 < 0 ? 0 : (k1 > 4 ? 4 : k1);
            const int wi = k0 + 5 * k1;
            msg += bs * base[(size_t)wi * CCH];
        }
    }
    atomicMax(&agg[(size_t)dst * CCH + c], enc_f32(msg));
}

// ---------------------------------------------------------------------------
// Finalize: relu(decode(agg) + x@root + bias). Sentinel -> 0 (isneginf path).
// ---------------------------------------------------------------------------
__global__ void finalize_kernel(
    const unsigned* __restrict__ agg, const float* __restrict__ XR,
    const float* __restrict__ bias, float* __restrict__ xout, int n)
{
    int i = blockIdx.x * blockDim.x + threadIdx.x;
    if (i >= n) return;
    const unsigned key = agg[i];
    float a = 0.0f;
    if (key != 0u) {
        const unsigned u = (key & 0x80000000u) ? (key & 0x7FFFFFFFu) : ~key;
        a = __uint_as_float(u);
    }
    const float v = a + XR[i] + bias[i & (CCH - 1)];
    xout[i] = v > 0.0f ? v : 0.0f;
}

// ---------------------------------------------------------------------------
// Orchestration. Workspace layout (floats):
//   XW  : N*25*64  (320 MB, reused across the two layers)
//   XR  : N*64
//   AGG : N*64 (uint)
//   x1  : N*64
//   x2  : N*64
// ---------------------------------------------------------------------------
extern "C" void kernel_launch(void* const* d_in, const int* in_sizes, int n_in,
                              void* d_out, int out_size, void* d_ws, size_t ws_size,
                              hipStream_t stream)
{
    const float* x    = (const float*)d_in[0];
    const int*   ei   = (const int*)  d_in[1];
    const float* attr = (const float*)d_in[2];
    const float* W1   = (const float*)d_in[3];
    const float* r1   = (const float*)d_in[4];
    const float* b1   = (const float*)d_in[5];
    const float* W2   = (const float*)d_in[6];
    const float* r2   = (const float*)d_in[7];
    const float* b2   = (const float*)d_in[8];
    const float* fw   = (const float*)d_in[9];
    const float* fb   = (const float*)d_in[10];
    float*       out  = (float*)d_out;

    const int N = in_sizes[0] / CCH;    // 50000
    const int E = in_sizes[1] / 2;      // 800000
    const int ntiles = N / 16;          // 3125 exact

    float*    XW  = (float*)d_ws;
    float*    XR  = XW + (size_t)N * NK * CCH;
    unsigned* AGG = (unsigned*)(XR + (size_t)N * CCH);
    float*    x1  = (float*)(AGG + (size_t)N * CCH);
    float*    x2  = x1 + (size_t)N * CCH;

    const int nc = N * CCH;
    for (int layer = 0; layer < 2; ++layer) {
        const float* xin  = layer == 0 ? x  : x1;
        const float* Wl   = layer == 0 ? W1 : W2;
        const float* Rl   = layer == 0 ? r1 : r2;
        const float* Bl   = layer == 0 ? b1 : b2;
        float*       xout = layer == 0 ? x1 : x2;

        xw_wmma_kernel<<<dim3(ntiles), 128, 0, stream>>>(xin, Wl, NK, XW, (long)(NK * CCH));
        xw_wmma_kernel<<<dim3(ntiles), 128, 0, stream>>>(xin, Rl, 1, XR, (long)CCH);
        init_agg_kernel<<<(nc + 255) / 256, 256, 0, stream>>>(AGG, nc);
        edge_kernel<<<(E + 3) / 4, 256, 0, stream>>>(XW, ei, attr, AGG, E);
        finalize_kernel<<<(nc + 255) / 256, 256, 0, stream>>>(AGG, XR, Bl, xout, nc);
    }
    final_wmma_kernel<<<dim3(ntiles), 128, 0, stream>>>(x, x1, x2, fw, fb, out);
}